// BlockDiagonalACDC_77893526880643
// MI455X (gfx1250) — compile-verified
//
#include <hip/hip_runtime.h>
#include <math.h>

// ---------------------------------------------------------------------------
// ACDC layer on MI455X (gfx1250), fp32 end-to-end via V_WMMA_F32_16X16X4_F32.
//
//   out = riffle( idct2( gconv( dct2( gconv(x, A) ), D) ) ) + bias
//
// Folded into:  CA  = blockdiag(A) @ C      (C  = unnormalized DCT-II matrix)
//               DCi = blockdiag(D) @ Cinv   (Cinv = matching iDCT matrix)
//               Y   = X @ CA                (dense 16384x4096x4096 GEMM)
//               OUT = riffle(Y @ DCi) + bias (riffle+bias fused in epilogue)
//
// The 64MB B-matrix of each GEMM stays L2-resident (192MB L2), so both GEMMs
// are HBM-streaming bound on X/Y/OUT. GEMM inner loop uses async global->LDS
// (ASYNCcnt-tracked GLOBAL_LOAD_ASYNC_TO_LDS_B128) double buffering with a
// single barrier per K-tile.
//
// ws layout: CA (64MB) | DCi (64MB) | Y (256MB)
// ---------------------------------------------------------------------------

#define NB     16384
#define DF     4096
#define NGRP   32
#define CPG    128
#define TWO_N  8192      // 2*DF
#define PERIOD 16384     // cos period in units of pi/TWO_N
#define PI_F   3.14159265358979323846f

typedef __attribute__((ext_vector_type(2))) float v2f;
typedef __attribute__((ext_vector_type(8))) float v8f;
typedef __attribute__((ext_vector_type(4))) int   v4i;

#define AS1 __attribute__((address_space(1)))
#define AS3 __attribute__((address_space(3)))

#if defined(__has_builtin)
#if __has_builtin(__builtin_amdgcn_global_load_async_to_lds_b128)
#define HAVE_ASYNC_LDS 1
#endif
#endif
#ifndef HAVE_ASYNC_LDS
#define HAVE_ASYNC_LDS 0
#endif

static __device__ __forceinline__ void wait_async_all() {
#if defined(__has_builtin) && __has_builtin(__builtin_amdgcn_s_wait_asynccnt)
    __builtin_amdgcn_s_wait_asynccnt(0);
#elif HAVE_ASYNC_LDS
    asm volatile("s_wait_asynccnt 0x0" ::: "memory");
#endif
}

// ---------------------------------------------------------------------------
// Build folded matrix M[j,k], j = g*128+i:
//   forward:  M[j,k] = sum_o W[g,o,i] * 2*cos(pi*k*(2*(g*128+o)+1)/(2N))
//   inverse:  M[j,k] = sum_o W[g,o,i] * Ci[g*128+o, k]
//             Ci[r,k] = (r==0) ? 1/(2N) : (1/N)*cos(pi*r*(2k+1)/(2N))
// Cos argument reduced EXACTLY: integer numerator mod 16384, angle in [0,2pi).
// grid: (NGRP, DF/128), block: 256
// ---------------------------------------------------------------------------
__global__ __launch_bounds__(256) void gen_folded(const float* __restrict__ W,
                                                  float* __restrict__ M,
                                                  int is_idct) {
    const int g  = blockIdx.x;
    const int kb = blockIdx.y * 128;
    const int t  = threadIdx.x;

    __shared__ float Wg[32][128];   // chunk of W: [oc][i]
    __shared__ float Cs[32][132];   // chunk of DCT rows: [oc][k], padded

    const int i0 = (t >> 4) * 8;    // 0,8,...,120
    const int k0 = (t & 15) * 8;    // 0,8,...,120

    float acc[8][8];
#pragma unroll
    for (int u = 0; u < 8; ++u)
#pragma unroll
        for (int v = 0; v < 8; ++v) acc[u][v] = 0.0f;

    for (int oc0 = 0; oc0 < CPG; oc0 += 32) {
        const float* wbase = W + ((size_t)g * CPG + oc0) * CPG;
        for (int e = t; e < 32 * 128; e += 256)
            Wg[e >> 7][e & 127] = wbase[e];
        for (int e = t; e < 32 * 128; e += 256) {
            const int oc = e >> 7;
            const int k  = kb + (e & 127);
            const int j  = g * CPG + oc0 + oc;   // global dct row
            float val;
            if (is_idct) {
                if (j == 0) {
                    val = 1.0f / (float)TWO_N;
                } else {
                    const int prod = j * (2 * k + 1);
                    val = (1.0f / (float)DF) *
                          cosf((float)(prod & (PERIOD - 1)) * (PI_F / (float)TWO_N));
                }
            } else {
                const int prod = k * (2 * j + 1);
                val = 2.0f * cosf((float)(prod & (PERIOD - 1)) * (PI_F / (float)TWO_N));
            }
            Cs[oc][e & 127] = val;
        }
        __syncthreads();
#pragma unroll 4
        for (int oc = 0; oc < 32; ++oc) {
            float a[8], c[8];
#pragma unroll
            for (int u = 0; u < 8; ++u) a[u] = Wg[oc][i0 + u];
#pragma unroll
            for (int v = 0; v < 8; ++v) c[v] = Cs[oc][k0 + v];
#pragma unroll
            for (int u = 0; u < 8; ++u)
#pragma unroll
                for (int v = 0; v < 8; ++v) acc[u][v] = fmaf(a[u], c[v], acc[u][v]);
        }
        __syncthreads();
    }

#pragma unroll
    for (int u = 0; u < 8; ++u) {
        const size_t row = (size_t)(g * CPG + i0 + u);
#pragma unroll
        for (int v = 0; v < 8; ++v)
            M[row * DF + kb + k0 + v] = acc[u][v];
    }
}

// ---------------------------------------------------------------------------
// Dense fp32 GEMM: OUT = X[NB,DF] @ B[DF,DF] via V_WMMA_F32_16X16X4_F32.
// Workgroup tile 128x128, BK=16, 8 waves, each wave owns a 16x128 strip
// (8 x 16x16 f32 accumulators). Async global->LDS double buffering.
// A in LDS as [m][k] stride 20 (conflict-free b64 frag loads).
// B in LDS as [k][n] stride 136 (async-copyable; 2*136 % 64 == 16 so the two
// half-wave K-rows of a fragment load hit disjoint bank sets).
// One barrier per K-tile:
//   wait_async(my writes of buf landed) ; barrier (everyone's landed AND
//   everyone finished reading buf^1) ; prefetch buf^1 ; compute from buf.
// grid: (NB/128, DF/128), block: 256
// ---------------------------------------------------------------------------
#define BM 128
#define BN 128
#define BK 16
#define LDA 20    // A row stride (floats)
#define LDB 136   // B row stride (floats)

__global__ __launch_bounds__(256) void gemm_wmma_f32(const float* __restrict__ X,
                                                     const float* __restrict__ B,
                                                     float* __restrict__ OUT,
                                                     const float* __restrict__ bias,
                                                     int do_riffle) {
    __shared__ float Asl[2][BM][LDA];   // [buf][m][k]
    __shared__ float Bsl[2][BK][LDB];   // [buf][k][n]

    const int t     = threadIdx.x;
    const int lane  = t & 31;
    const int wave  = t >> 5;            // 0..7
    const int m0    = blockIdx.x * BM;
    const int n0    = blockIdx.y * BN;
    const int lrow  = lane & 15;         // row/col within 16
    const int khalf = (lane >> 4) << 1;  // 0 or 2: K-pair selector (ISA A/B layout)
    const int wm    = wave * 16;         // wave's M offset inside tile

    // per-thread tile-staging coordinates (16B chunks)
    const int ar  = t >> 2;              // A: row 0..63
    const int ac  = (t & 3) * 4;         // A: col 0,4,8,12
    const int br0 = t >> 5;              // B: rows 0..7 / +8 for second chunk
    const int bc  = (t & 31) * 4;        // B: col 0..124

    v8f acc[8];
#pragma unroll
    for (int nt = 0; nt < 8; ++nt)
#pragma unroll
        for (int r = 0; r < 8; ++r) acc[nt][r] = 0.0f;

    // ---- stage one K-tile (kt) into buffer buf ----
    auto issue_tile = [&](int kt, int buf) {
#if HAVE_ASYNC_LDS
        __builtin_amdgcn_global_load_async_to_lds_b128(
            (AS1 v4i*)(X + (size_t)(m0 + ar) * DF + kt + ac),
            (AS3 v4i*)&Asl[buf][ar][ac], 0, 0);
        __builtin_amdgcn_global_load_async_to_lds_b128(
            (AS1 v4i*)(B + (size_t)(kt + br0) * DF + n0 + bc),
            (AS3 v4i*)&Bsl[buf][br0][bc], 0, 0);
        __builtin_amdgcn_global_load_async_to_lds_b128(
            (AS1 v4i*)(B + (size_t)(kt + br0 + 8) * DF + n0 + bc),
            (AS3 v4i*)&Bsl[buf][br0 + 8][bc], 0, 0);
#else
        const float4 va = *(const float4*)(X + (size_t)(m0 + ar) * DF + kt + ac);
        *(float4*)&Asl[buf][ar][ac] = va;
        const float4 vb0 = *(const float4*)(B + (size_t)(kt + br0) * DF + n0 + bc);
        *(float4*)&Bsl[buf][br0][bc] = vb0;
        const float4 vb1 = *(const float4*)(B + (size_t)(kt + br0 + 8) * DF + n0 + bc);
        *(float4*)&Bsl[buf][br0 + 8][bc] = vb1;
#endif
    };

    issue_tile(0, 0);

    const int NT = DF / BK;   // 256 K-tiles
    for (int i = 0; i < NT; ++i) {
        const int buf = i & 1;
        // data for `buf` was issued last iteration (or prologue): wait + converge.
        // This barrier also proves every wave finished reading buf^1 (program
        // order: compute(i-1) precedes this barrier), so refilling it is safe.
        wait_async_all();
        __syncthreads();
        if (i + 1 < NT) issue_tile((i + 1) * BK, buf ^ 1);

#pragma unroll
        for (int kk = 0; kk < BK; kk += 4) {
            // A frag (16x4): lanes 0-15 hold K=kk,kk+1; lanes 16-31 K=kk+2,kk+3
            const v2f a = *(const v2f*)&Asl[buf][wm + lrow][kk + khalf];
#pragma unroll
            for (int nt = 0; nt < 8; ++nt) {
                const int n = nt * 16 + lrow;
                v2f b;
                b.x = Bsl[buf][kk + khalf][n];
                b.y = Bsl[buf][kk + khalf + 1][n];
                acc[nt] = __builtin_amdgcn_wmma_f32_16x16x4_f32(
                    /*neg_a=*/false, a, /*neg_b=*/false, b,
                    /*c_mod=*/(short)0, acc[nt],
                    /*reuse_a=*/false, /*reuse_b=*/false);
            }
        }
    }

    // ---- epilogue: C/D layout: vgpr r -> row r (lanes<16) / r+8 (lanes>=16) ----
    const int rbase = m0 + wm + ((lane >> 4) << 3);
#pragma unroll
    for (int nt = 0; nt < 8; ++nt) {
        const int col = n0 + nt * 16 + lrow;
        int   ocol = col;
        float bv   = 0.0f;
        if (do_riffle) {
            // riffle: out[:, (c&1)*2048 + (c>>1)] = y[:, c]; bias indexed post-riffle
            ocol = ((col & 1) ? (DF / 2) : 0) + (col >> 1);
            bv   = bias[ocol];
        }
#pragma unroll
        for (int r = 0; r < 8; ++r)
            OUT[(size_t)(rbase + r) * DF + ocol] = acc[nt][r] + bv;
    }
}

// ---------------------------------------------------------------------------
extern "C" void kernel_launch(void* const* d_in, const int* in_sizes, int n_in,
                              void* d_out, int out_size, void* d_ws, size_t ws_size,
                              hipStream_t stream) {
    const float* x    = (const float*)d_in[0];   // [16384, 4096]
    const float* A    = (const float*)d_in[1];   // [32, 128, 128]
    const float* D    = (const float*)d_in[2];   // [32, 128, 128]
    const float* bias = (const float*)d_in[3];   // [1, 4096]
    float* out = (float*)d_out;

    float* CA  = (float*)d_ws;                       //  64 MB
    float* DCi = CA  + (size_t)DF * DF;              //  64 MB
    float* Y   = DCi + (size_t)DF * DF;              // 256 MB

    dim3 gGen(NGRP, DF / 128);
    gen_folded<<<gGen, 256, 0, stream>>>(A, CA, 0);
    gen_folded<<<gGen, 256, 0, stream>>>(D, DCi, 1);

    dim3 gGemm(NB / BM, DF / BN);
    gemm_wmma_f32<<<gGemm, 256, 0, stream>>>(x, CA, Y, bias, 0);
    gemm_wmma_f32<<<gGemm, 256, 0, stream>>>(Y, DCi, out, bias, 1);
}